// EmbeddingProposal_17695265260041
// MI455X (gfx1250) — compile-verified
//
#include <hip/hip_runtime.h>
#include <hip/hip_bf16.h>
#include <stdint.h>

// ---------------------------------------------------------------------------
// Problem constants (match reference): K=256 trees, T=256 leaves, D=64, H=256
// ---------------------------------------------------------------------------
#define KK 256
#define TT 256
#define DD 64
#define HH 256
#define LDS_STRIDE 68   // 64 + 4 pad: conflict-free float2 column reads

#define GTAB_ELEMS ((size_t)KK * TT * TT)       // 2^24 gumbels
#define GTAB_BYTES (GTAB_ELEMS * sizeof(float)) // 64 MB
#define WS_GTAB_OFF 4096                        // idx/lse live below this

typedef float v2f __attribute__((ext_vector_type(2)));
typedef float v8f __attribute__((ext_vector_type(8)));

// ---------------------------------------------------------------------------
// Threefry-2x32 (20 rounds), key = jax.random.key(42) -> (k0,k1) = (0,42)
// ---------------------------------------------------------------------------
__device__ __forceinline__ uint32_t rotl32(uint32_t x, int r) {
  return (x << r) | (x >> (32 - r));
}

__device__ __forceinline__ uint2 threefry2x32_42(uint32_t x0, uint32_t x1) {
  const uint32_t k0 = 0u, k1 = 42u;
  const uint32_t k2 = 0x1BD11BDAu ^ k0 ^ k1;
  x0 += k0; x1 += k1;
#define TF_R4(a, b, c, d)                                     \
  x0 += x1; x1 = rotl32(x1, a); x1 ^= x0;                     \
  x0 += x1; x1 = rotl32(x1, b); x1 ^= x0;                     \
  x0 += x1; x1 = rotl32(x1, c); x1 ^= x0;                     \
  x0 += x1; x1 = rotl32(x1, d); x1 ^= x0;
  TF_R4(13, 15, 26, 6)  x0 += k1; x1 += k2 + 1u;
  TF_R4(17, 29, 16, 24) x0 += k2; x1 += k0 + 2u;
  TF_R4(13, 15, 26, 6)  x0 += k0; x1 += k1 + 3u;
  TF_R4(17, 29, 16, 24) x0 += k1; x1 += k2 + 4u;
  TF_R4(13, 15, 26, 6)  x0 += k2; x1 += k0 + 5u;
#undef TF_R4
  uint2 r; r.x = x0; r.y = x1; return r;
}

// JAX: uniform = bitcast((bits>>9)|0x3F800000) - 1, minval=tiny; gumbel = -log(-log(u))
__device__ __forceinline__ float bits_to_gumbel(uint32_t bits) {
  float u = __uint_as_float((bits >> 9) | 0x3F800000u) - 1.0f;
  u = fmaxf(u, 1.17549435e-38f);
  return -logf(-logf(u));
}

// ---------------------------------------------------------------------------
// Kernel 0 (table path): one threefry per counter pair, both halves used.
// Halves the threefry + logf work vs. inline generation.
// p in [0, 2^23), pairs (p, p + 2^23); out = concat(o0, o1)  (JAX layout).
// ---------------------------------------------------------------------------
__global__ __launch_bounds__(256) void k_gumbel_fill(float* __restrict__ gtab) {
  const uint32_t p = (uint32_t)blockIdx.x * 256u + (uint32_t)threadIdx.x;
  const uint2 r = threefry2x32_42(p, p + 0x800000u);
  gtab[p]             = bits_to_gumbel(r.x);
  gtab[p + 0x800000u] = bits_to_gumbel(r.y);
}

// ---------------------------------------------------------------------------
// Kernel 1: per-k fused  gram (f32 WMMA) -> dist -> logw -> gumbel-argmax,
// logsumexp.  One block (8 waves) per k.
// UseTab=true : gumbels read from precomputed table (L2-resident).
// UseTab=false: gumbels generated inline (discards half of each threefry).
// ---------------------------------------------------------------------------
template <bool UseTab>
__global__ __launch_bounds__(256) void k_pairwise_sample(
    const float* __restrict__ emb,   // [K,T,D]
    int* __restrict__ ws_i1, int* __restrict__ ws_i2,
    float* __restrict__ ws_lse, const float* __restrict__ gtab) {
  const int k   = blockIdx.x;
  const int tid = threadIdx.x;

  __shared__ float sE[TT * LDS_STRIDE];   // 69632 B
  __shared__ float s_sq[TT];
  __shared__ float s_red_v[256];
  __shared__ int   s_red_i[256];
  __shared__ float s_red_s[256];

  const float* E = emb + (size_t)k * TT * DD;

  // ---- cooperative, coalesced load: 256x64 f32 as float4s into padded LDS
  for (int i = tid; i < TT * (DD / 4); i += 256) {
    const float4 v = ((const float4*)E)[i];
    const int r = i >> 4;             // row
    const int c = (i & 15) << 2;      // col (multiple of 4)
    *(float4*)&sE[r * LDS_STRIDE + c] = v;
  }
  __syncthreads();

  // ---- per-row squared norms
  {
    float acc = 0.0f;
    const float* row = &sE[tid * LDS_STRIDE];
#pragma unroll
    for (int c = 0; c < DD; ++c) acc += row[c] * row[c];
    s_sq[tid] = acc;
  }
  __syncthreads();

  // ---- WMMA fragment addressing (V_WMMA_F32_16X16X4_F32, wave32)
  // A 16x4: lane<16 -> M=lane, K={0,1}; lane>=16 -> M=lane-16, K={2,3}
  // B 4x16: lane<16 -> N=lane, K={0,1}; lane>=16 -> N=lane-16, K={2,3}
  const int lane = tid & 31;
  const int wave = tid >> 5;
  const int rsub = lane & 15;
  const int kk2  = (lane >> 4) << 1;       // 0 or 2
  const int mnb  = (lane >> 4) << 3;       // D-frag: M = v + 8*(lane>=16)

  const int       halfsel = (k >= 128);    // which threefry output half we own
  const uint32_t  pbase   = (uint32_t)k << 16;

  float best    = -INFINITY;
  int   bestIdx = 0x7FFFFFFF;
  float sumExp  = 0.0f;

  for (int tile = wave; tile < 256; tile += 8) {
    const int ti = tile >> 4;
    const int tj = tile & 15;
    const int ra = ti * 16 + rsub;   // A rows
    const int rb = tj * 16 + rsub;   // B "rows" (columns of gram)

    v8f c = {0.0f, 0.0f, 0.0f, 0.0f, 0.0f, 0.0f, 0.0f, 0.0f};
#pragma unroll
    for (int ks = 0; ks < DD / 4; ++ks) {
      const int col = (ks << 2) + kk2;     // even -> 8B aligned in LDS
      v2f a = *(const v2f*)&sE[ra * LDS_STRIDE + col];
      v2f b = *(const v2f*)&sE[rb * LDS_STRIDE + col];
      c = __builtin_amdgcn_wmma_f32_16x16x4_f32(
          /*neg_a=*/false, a, /*neg_b=*/false, b,
          /*c_mod=*/(short)0, c, /*reuse_a=*/false, /*reuse_b=*/false);
    }

    // D layout: lane<16 -> N=lane, M=v; lane>=16 -> N=lane-16, M=v+8
    const int j   = tj * 16 + rsub;
    const float sqj = s_sq[j];
#pragma unroll
    for (int v = 0; v < 8; ++v) {
      const int i  = ti * 16 + mnb + v;
      float d2   = s_sq[i] + sqj - 2.0f * c[v];
      float logw = -sqrtf(fmaxf(d2, 0.0f));
      if (i == j) logw = -INFINITY;

      const int l = (i << 8) | j;                  // flat index i*T + j
      float g;
      if (UseTab) {
        g = gtab[pbase + (uint32_t)l];
      } else {
        const uint32_t p  = pbase + (uint32_t)l;
        const uint32_t c0 = halfsel ? (p - 0x800000u) : p;
        const uint2 r = threefry2x32_42(c0, c0 + 0x800000u);
        g = bits_to_gumbel(halfsel ? r.y : r.x);
      }

      sumExp += expf(logw);
      const float val = logw + g;
      if (val > best || (val == best && l < bestIdx)) { best = val; bestIdx = l; }
    }
  }

  // ---- block reduction: (max, lowest-index tiebreak) and sum
  s_red_v[tid] = best; s_red_i[tid] = bestIdx; s_red_s[tid] = sumExp;
  __syncthreads();
  for (int off = 128; off > 0; off >>= 1) {
    if (tid < off) {
      const float ov = s_red_v[tid + off];
      const int   oi = s_red_i[tid + off];
      if (ov > s_red_v[tid] || (ov == s_red_v[tid] && oi < s_red_i[tid])) {
        s_red_v[tid] = ov; s_red_i[tid] = oi;
      }
      s_red_s[tid] += s_red_s[tid + off];
    }
    __syncthreads();
  }
  if (tid == 0) {
    const int fi = s_red_i[0];
    ws_i1[k]  = fi >> 8;
    ws_i2[k]  = fi & 255;
    ws_lse[k] = logf(s_red_s[0]);
  }
}

// ---------------------------------------------------------------------------
// Kernel 2: gather children, MLP, branch lengths, log probs, leaf counts.
// One block (256 threads) per k.  Output tuple, flat, in return order:
// [idx1 |256] [idx2 |256] [br1 |256] [br2 |256] [emb |256*64] [lvp |256] [lvm |256]
// ---------------------------------------------------------------------------
__global__ __launch_bounds__(256) void k_finish(
    const float* __restrict__ emb, const int* __restrict__ lc,
    const float* __restrict__ W1, const float* __restrict__ b1,
    const float* __restrict__ W2, const float* __restrict__ b2,
    const int* __restrict__ ws_i1, const int* __restrict__ ws_i2,
    const float* __restrict__ ws_lse, const int* __restrict__ Nptr,
    float* __restrict__ out) {
  const int k   = blockIdx.x;
  const int tid = threadIdx.x;

  __shared__ float c1[DD], c2[DD], h[HH], o[DD];
  __shared__ float s_red[2 * DD];
  __shared__ int   s_ones[256];

  const int i1 = ws_i1[k];
  const int i2 = ws_i2[k];
  const float* E = emb + (size_t)k * TT * DD;

  if (tid < DD)            c1[tid]      = E[i1 * DD + tid];
  else if (tid < 2 * DD)   c2[tid - DD] = E[i2 * DD + (tid - DD)];
  s_ones[tid] = (lc[k * TT + tid] == 1) ? 1 : 0;
  __syncthreads();

  // h = relu([c1,c2] @ W1 + b1) ; W1 is [128,256] row-major -> coalesced over tid
  {
    float acc = b1[tid];
#pragma unroll 4
    for (int i = 0; i < DD; ++i) acc += c1[i] * W1[i * HH + tid];
#pragma unroll 4
    for (int i = 0; i < DD; ++i) acc += c2[i] * W1[(DD + i) * HH + tid];
    h[tid] = fmaxf(acc, 0.0f);
  }
  __syncthreads();

  // out_emb = h @ W2 + b2 ; W2 is [256,64] row-major -> coalesced over tid
  if (tid < DD) {
    float acc = b2[tid];
#pragma unroll 4
    for (int j = 0; j < HH; ++j) acc += h[j] * W2[j * DD + tid];
    o[tid] = acc;
    out[4 * KK + k * DD + tid] = acc;
  }
  __syncthreads();

  if (tid < DD) {
    const float d1 = c1[tid] - o[tid];
    const float d2 = c2[tid] - o[tid];
    s_red[tid]      = d1 * d1;
    s_red[DD + tid] = d2 * d2;
  }
  __syncthreads();

  for (int off = 128; off > 0; off >>= 1) {
    if (tid < off) s_ones[tid] += s_ones[tid + off];
    __syncthreads();
  }

  if (tid == 0) {
    float b1s = 0.0f, b2s = 0.0f;
    for (int d = 0; d < DD; ++d) { b1s += s_red[d]; b2s += s_red[DD + d]; }
    out[2 * KK + k] = sqrtf(b1s);
    out[3 * KK + k] = sqrtf(b2s);

    // log merge prob: dist(i1,i2) via same sq_i+sq_j-2*gram formula
    float s1 = 0.0f, s2 = 0.0f, g12 = 0.0f;
    for (int d = 0; d < DD; ++d) {
      const float a = c1[d], b = c2[d];
      s1 += a * a; s2 += b * b; g12 += a * b;
    }
    const float dist = sqrtf(fmaxf(s1 + s2 - 2.0f * g12, 0.0f));
    out[4 * KK + KK * DD + k] = -dist + 0.69314718055994530942f - ws_lse[k];

    const int ones = s_ones[0] - ((lc[k * TT + i1] == 1) ? 1 : 0)
                               - ((lc[k * TT + i2] == 1) ? 1 : 0);
    out[5 * KK + KK * DD + k] = logf((float)(Nptr[0] - ones));

    out[k]      = (float)i1;
    out[KK + k] = (float)i2;
  }
}

// ---------------------------------------------------------------------------
// Inputs (setup_inputs order):
//  0:N(int,1) 1:leaf_counts(int,K*T) 2:embeddings(f32,K*T*D) 3:log(int,1)
//  4:W1(f32,128*256) 5:b1(f32,256) 6:W2(f32,256*64) 7:b2(f32,64)
// Workspace: [0,4096): ws_i1|ws_i2|ws_lse ; [4096, 4096+64MB): gumbel table
// (table path only taken when ws_size is large enough; choice is a pure
// function of ws_size, so behavior is deterministic across calls).
// ---------------------------------------------------------------------------
extern "C" void kernel_launch(void* const* d_in, const int* in_sizes, int n_in,
                              void* d_out, int out_size, void* d_ws, size_t ws_size,
                              hipStream_t stream) {
  const int*   dN  = (const int*)d_in[0];
  const int*   lc  = (const int*)d_in[1];
  const float* emb = (const float*)d_in[2];
  const float* W1  = (const float*)d_in[4];
  const float* b1  = (const float*)d_in[5];
  const float* W2  = (const float*)d_in[6];
  const float* b2  = (const float*)d_in[7];
  float* out = (float*)d_out;

  int*   ws_i1  = (int*)d_ws;
  int*   ws_i2  = ws_i1 + KK;
  float* ws_lse = (float*)(ws_i2 + KK);
  float* gtab   = (float*)((char*)d_ws + WS_GTAB_OFF);

  const bool useTab = ws_size >= (size_t)WS_GTAB_OFF + GTAB_BYTES;

  if (useTab) {
    // 2^23 counter pairs, 256 threads each -> 32768 blocks
    k_gumbel_fill<<<32768, 256, 0, stream>>>(gtab);
    k_pairwise_sample<true><<<KK, 256, 0, stream>>>(emb, ws_i1, ws_i2, ws_lse, gtab);
  } else {
    k_pairwise_sample<false><<<KK, 256, 0, stream>>>(emb, ws_i1, ws_i2, ws_lse, nullptr);
  }
  k_finish<<<KK, 256, 0, stream>>>(emb, lc, W1, b1, W2, b2,
                                   ws_i1, ws_i2, ws_lse, dN, out);
}